// GaussianAdapter_70557722738880
// MI455X (gfx1250) — compile-verified
//
#include <hip/hip_runtime.h>
#include <math.h>

typedef __attribute__((ext_vector_type(2))) float v2f;
typedef __attribute__((ext_vector_type(8))) float v8f;

#define CCH     82      // channels per gaussian: 3 scale + 4 quat + 75 sh
#define KPAD    28      // 25 -> 28 (7 k-steps of 4)
#define MPAD    32      // 25 -> 32 (2 m-tiles of 16)
#define VSTRIDE 1024    // floats of workspace per view
#define DOFF    128     // D matrix offset within a view's workspace

// ---------------------------------------------------------------- SH basis
__device__ __forceinline__ void sh_basis25(float x, float y, float z, float* Y) {
  float x2 = x*x, y2 = y*y, z2 = z*z;
  Y[0]  = 0.2820947918f;
  Y[1]  = 0.4886025119f*y;  Y[2] = 0.4886025119f*z;  Y[3] = 0.4886025119f*x;
  Y[4]  = 1.0925484306f*x*y;
  Y[5]  = 1.0925484306f*y*z;
  Y[6]  = 0.3153915653f*(3.f*z2 - 1.f);
  Y[7]  = 1.0925484306f*x*z;
  Y[8]  = 0.5462742153f*(x2 - y2);
  Y[9]  = 0.5900435899f*y*(3.f*x2 - y2);
  Y[10] = 2.8906114426f*x*y*z;
  Y[11] = 0.4570457995f*y*(5.f*z2 - 1.f);
  Y[12] = 0.3731763326f*z*(5.f*z2 - 3.f);
  Y[13] = 0.4570457995f*x*(5.f*z2 - 1.f);
  Y[14] = 1.4453057213f*z*(x2 - y2);
  Y[15] = 0.5900435899f*x*(x2 - 3.f*y2);
  Y[16] = 2.5033429418f*x*y*(x2 - y2);
  Y[17] = 1.7701307698f*y*z*(3.f*x2 - y2);
  Y[18] = 0.9461746958f*x*y*(7.f*z2 - 1.f);
  Y[19] = 0.6690465436f*y*z*(7.f*z2 - 3.f);
  Y[20] = 0.1057855469f*(35.f*z2*z2 - 30.f*z2 + 3.f);
  Y[21] = 0.6690465436f*x*z*(7.f*z2 - 3.f);
  Y[22] = 0.4730873479f*(x2 - y2)*(7.f*z2 - 1.f);
  Y[23] = 1.7701307698f*x*z*(x2 - 3.f*y2);
  Y[24] = 0.6258357354f*(x2*x2 - 6.f*x2*y2 + y2*y2);
}

// ------------------------------------------------- per-view setup (8 blocks)
// Header (17 floats): R_c2w[9], cam_o[3], 1/fx, 1/fy, cx, cy, scale_adj
// D_pad[32][28] row-major at DOFF (padded block-diagonal SH rotation matrix).
__global__ void view_setup_kernel(const float* __restrict__ extr,
                                  const float* __restrict__ intr,
                                  const int*   __restrict__ imgh,
                                  const int*   __restrict__ imgw,
                                  float* __restrict__ ws) {
  const int v = blockIdx.x, t = threadIdx.x;
  float* W = ws + (size_t)v*VSTRIDE;
  const float* E = extr + (size_t)v*16;
  const float* K = intr + (size_t)v*9;

  for (int i = t; i < MPAD*KPAD; i += 32) W[DOFF + i] = 0.f;
  __syncthreads();

  float Rw[3][3];   // R_w2c
  #pragma unroll
  for (int i = 0; i < 3; i++)
    #pragma unroll
    for (int j = 0; j < 3; j++) Rw[i][j] = E[i*4 + j];

  if (t == 0) {
    #pragma unroll
    for (int i = 0; i < 3; i++)
      #pragma unroll
      for (int j = 0; j < 3; j++) W[i*3 + j] = Rw[j][i];   // R_c2w = Rᵀ
    float tx = E[3], ty = E[7], tz = E[11];
    #pragma unroll
    for (int i = 0; i < 3; i++)
      W[9 + i] = -(Rw[0][i]*tx + Rw[1][i]*ty + Rw[2][i]*tz); // cam_o = -Rᵀt
    float fx = K[0], fy = K[4], cx = K[2], cy = K[5];
    W[12] = 1.f/fx; W[13] = 1.f/fy; W[14] = cx; W[15] = cy;
    float iw = 1.f/(float)imgw[0], ih = 1.f/(float)imgh[0];
    W[16] = 0.1f*(iw/fx + ih/fy);          // SCALE_PCT * einsum(K2inv, px_size)
    W[DOFF + 0] = 1.f;                     // degree-0 band: identity
  }

  // Threads 0..3: band l = t+1. D_l = B0^{-1} * Brot (sample-independent
  // rotation representation; Fibonacci-sphere samples give invertible B0).
  if (t < 4) {
    const int l = t + 1, nb = 2*l + 1, lo = l*l;
    float A[9][18];     // augmented [B0 | Brot]
    float Y[25];
    for (int i = 0; i < nb; i++) {
      float zz = 1.f - 2.f*((float)i + 0.5f)/(float)nb;
      float rr = sqrtf(fmaxf(0.f, 1.f - zz*zz));
      float ph = 2.39996323f*(float)i + 0.1f*(float)l;
      float sx = rr*cosf(ph), sy = rr*sinf(ph), sz = zz;
      sh_basis25(sx, sy, sz, Y);
      for (int k = 0; k < nb; k++) A[i][k] = Y[lo + k];
      float dx = Rw[0][0]*sx + Rw[0][1]*sy + Rw[0][2]*sz;   // d = R_w2c·S
      float dy = Rw[1][0]*sx + Rw[1][1]*sy + Rw[1][2]*sz;
      float dz = Rw[2][0]*sx + Rw[2][1]*sy + Rw[2][2]*sz;
      sh_basis25(dx, dy, dz, Y);
      for (int k = 0; k < nb; k++) A[i][nb + k] = Y[lo + k];
    }
    for (int p = 0; p < nb; p++) {           // Gauss-Jordan, partial pivot
      int piv = p; float best = fabsf(A[p][p]);
      for (int i = p + 1; i < nb; i++) {
        float a = fabsf(A[i][p]); if (a > best) { best = a; piv = i; }
      }
      if (piv != p)
        for (int j = 0; j < 2*nb; j++) { float tm = A[p][j]; A[p][j] = A[piv][j]; A[piv][j] = tm; }
      float inv = 1.f/A[p][p];
      for (int j = 0; j < 2*nb; j++) A[p][j] *= inv;
      for (int i = 0; i < nb; i++) if (i != p) {
        float f = A[i][p];
        for (int j = 0; j < 2*nb; j++) A[i][j] -= f*A[p][j];
      }
    }
    for (int q = 0; q < nb; q++)
      for (int k = 0; k < nb; k++)
        W[DOFF + (lo + q)*KPAD + (lo + k)] = A[q][nb + k];
  }
}

// --------------------------------------------------------- fused main kernel
// One wave owns 32 gaussians: lane l does the elementwise geometry of
// gaussian g0+l (full EXEC, no divergence), then the SH rotation runs as a
// WMMA GEMM  out(25x32) = D(25x25[pad 32x28]) * in(25x32)  split into two
// 16-column N-tiles: 2 m-tiles x 7 k-steps x 2 n-tiles x 3 colors = 84
// V_WMMA_F32_16X16X4_F32 per wave. pg is streamed exactly once.
__global__ void __launch_bounds__(256)
adapter_kernel(const float* __restrict__ pg, const float* __restrict__ pc,
               const float* __restrict__ opac, const float* __restrict__ dep,
               const float* __restrict__ ws, float* __restrict__ out,
               int Ntot, int Rview) {
  const int lane = threadIdx.x & 31;
  const int wave = blockIdx.x*(blockDim.x >> 5) + (threadIdx.x >> 5);
  const int g0 = wave*32;
  if (g0 >= Ntot) return;                        // uniform per wave
  const size_t N = (size_t)Ntot;
  const float* W = ws + (size_t)(g0 / Rview)*VSTRIDE;   // view header (wave-uniform)

  // ---------------- per-lane geometry for gaussian g0+lane ----------------
  {
    int g = g0 + lane; if (g >= Ntot) g = Ntot - 1;     // clamp (stores masked)
    const bool valid = (g0 + lane) < Ntot;
    float Rc[3][3];
    #pragma unroll
    for (int i = 0; i < 9; i++) Rc[i/3][i%3] = W[i];
    float ox = W[9], oy = W[10], oz = W[11];
    float ifx = W[12], ify = W[13], cx = W[14], cy = W[15], sadj = W[16];

    const float* G = pg + (size_t)g*CCH;
    float sr0 = G[0], sr1 = G[1], sr2 = G[2];
    float qw = G[3], qx = G[4], qy = G[5], qz = G[6];
    float px = pc[(size_t)g*2 + 0], py = pc[(size_t)g*2 + 1];
    float op = opac[g], d = dep[g];

    float sc = d*sadj;
    float s0 = (0.01f + 99.99f/(1.f + __expf(-sr0)))*sc;
    float s1 = (0.01f + 99.99f/(1.f + __expf(-sr1)))*sc;
    float s2 = (0.01f + 99.99f/(1.f + __expf(-sr2)))*sc;

    float qn = rsqrtf(qw*qw + qx*qx + qy*qy + qz*qz);
    qw *= qn; qx *= qn; qy *= qn; qz *= qn;
    float Rq[3][3] = {
      {1.f-2.f*(qy*qy+qz*qz), 2.f*(qx*qy-qw*qz),     2.f*(qx*qz+qw*qy)},
      {2.f*(qx*qy+qw*qz),     1.f-2.f*(qx*qx+qz*qz), 2.f*(qy*qz-qw*qx)},
      {2.f*(qx*qz-qw*qy),     2.f*(qy*qz+qw*qx),     1.f-2.f*(qx*qx+qy*qy)}};
    float M[3][3];
    #pragma unroll
    for (int i = 0; i < 3; i++)
      #pragma unroll
      for (int j = 0; j < 3; j++)
        M[i][j] = Rc[i][0]*Rq[0][j] + Rc[i][1]*Rq[1][j] + Rc[i][2]*Rq[2][j];
    float ss[3] = {s0*s0, s1*s1, s2*s2};

    float dx = (px - cx)*ifx, dy = (py - cy)*ify;
    float rn = rsqrtf(dx*dx + dy*dy + 1.f);
    dx *= rn; dy *= rn; float dz = rn;
    float rdx = Rc[0][0]*dx + Rc[0][1]*dy + Rc[0][2]*dz;
    float rdy = Rc[1][0]*dx + Rc[1][1]*dy + Rc[1][2]*dz;
    float rdz = Rc[2][0]*dx + Rc[2][1]*dy + Rc[2][2]*dz;

    if (valid) {
      float* means  = out;                 // [N][3]
      float* scales = out + 3*N;           // [N][3]
      float* cov    = out + 6*N;           // [N][9]
      float* quats  = out + 15*N;          // [N][4]
      float* opo    = out + 19*N;          // [N]
      means[(size_t)g*3+0] = ox + rdx*d;
      means[(size_t)g*3+1] = oy + rdy*d;
      means[(size_t)g*3+2] = oz + rdz*d;
      scales[(size_t)g*3+0] = s0; scales[(size_t)g*3+1] = s1; scales[(size_t)g*3+2] = s2;
      #pragma unroll
      for (int i = 0; i < 3; i++)
        #pragma unroll
        for (int j = 0; j < 3; j++)
          cov[(size_t)g*9 + i*3 + j] =
            M[i][0]*ss[0]*M[j][0] + M[i][1]*ss[1]*M[j][1] + M[i][2]*ss[2]*M[j][2];
      quats[(size_t)g*4+0] = qw; quats[(size_t)g*4+1] = qx;
      quats[(size_t)g*4+2] = qy; quats[(size_t)g*4+3] = qz;
      opo[g] = op;
    }
  }

  // ------------------------------ SH rotation -----------------------------
  const int n = lane & 15, kh = lane >> 4;
  const float khm = kh ? 0.f : 1.f;              // mask for k-step 6 (no branch)
  const float* Dp = W + DOFF;
  float* out_sh = out + 20*N;

  // A operand (16x4 f32): lane = row M, vgpr pair holds K = 2*kh, 2*kh+1
  v2f a[2][7];
  #pragma unroll
  for (int mt = 0; mt < 2; mt++)
    #pragma unroll
    for (int ks = 0; ks < 7; ks++) {
      int m = mt*16 + n, k = ks*4 + 2*kh;
      a[mt][ks].x = Dp[m*KPAD + k];
      a[mt][ks].y = Dp[m*KPAD + k + 1];
    }

  #pragma unroll
  for (int t = 0; t < 2; t++) {                  // two 16-column N-tiles
    int gc = g0 + t*16 + n;
    const bool valid = gc < Ntot;
    if (gc >= Ntot) gc = Ntot - 1;               // clamp loads; stores masked
    const float* X = pg + (size_t)gc*CCH + 7;
    #pragma unroll
    for (int c = 0; c < 3; c++) {
      const float* xc = X + c*25;
      v8f acc0 = {}; v8f acc1 = {};
      #pragma unroll
      for (int ks = 0; ks < 6; ks++) {           // k = 0..23: always in-band
        int k = ks*4 + 2*kh;
        v2f b; b.x = xc[k]; b.y = xc[k + 1];     // B (4x16): lane = col N
        acc0 = __builtin_amdgcn_wmma_f32_16x16x4_f32(false, a[0][ks], false, b,
                                                     (short)0, acc0, false, false);
        acc1 = __builtin_amdgcn_wmma_f32_16x16x4_f32(false, a[1][ks], false, b,
                                                     (short)0, acc1, false, false);
      }
      {                                          // k-step 6: only k=24 is live;
        v2f b; b.x = xc[24]*khm; b.y = 0.f;      // unconditional load + mask,
        acc0 = __builtin_amdgcn_wmma_f32_16x16x4_f32(false, a[0][6], false, b,
                                                     (short)0, acc0, false, false);
        acc1 = __builtin_amdgcn_wmma_f32_16x16x4_f32(false, a[1][6], false, b,
                                                     (short)0, acc1, false, false);
      }
      // C/D layout: vgpr j -> row 8*kh + j (tile 0) / 16 + 8*kh + j (tile 1)
      if (valid) {
        float* o = out_sh + ((size_t)gc*3 + c)*25;
        const int mb = 8*kh;
        #pragma unroll
        for (int j = 0; j < 8; j++) o[mb + j] = acc0[j];
        #pragma unroll
        for (int j = 0; j < 8; j++) {
          int m = 16 + mb + j;
          if (m < 25) o[m] = acc1[j];
        }
      }
    }
  }
}

// ---------------------------------------------------------------- launcher
extern "C" void kernel_launch(void* const* d_in, const int* in_sizes, int n_in,
                              void* d_out, int out_size, void* d_ws, size_t ws_size,
                              hipStream_t stream) {
  const float* pg   = (const float*)d_in[0];
  const float* pc   = (const float*)d_in[1];
  const float* extr = (const float*)d_in[2];
  const float* intr = (const float*)d_in[3];
  const float* opac = (const float*)d_in[4];
  const float* dep  = (const float*)d_in[5];
  const int*   imgh = (const int*)d_in[6];
  const int*   imgw = (const int*)d_in[7];

  const int N  = in_sizes[4];          // B*V*R gaussians
  const int NV = in_sizes[2] / 16;     // B*V views
  const int Rv = N / NV;               // gaussians per view

  float* ws  = (float*)d_ws;
  float* out = (float*)d_out;

  view_setup_kernel<<<NV, 32, 0, stream>>>(extr, intr, imgh, imgw, ws);
  const int waves  = (N + 31)/32;      // 32 gaussians per wave
  const int blocks = (waves + 7)/8;    // 8 waves (256 threads) per block
  adapter_kernel<<<blocks, 256, 0, stream>>>(pg, pc, opac, dep, ws, out, N, Rv);
}